// SimplicialConvolution_43198781063246
// MI455X (gfx1250) — compile-verified
//
#include <hip/hip_runtime.h>
#include <hip/hip_bf16.h>

typedef float v2f __attribute__((ext_vector_type(2)));
typedef float v4f __attribute__((ext_vector_type(4)));
typedef float v8f __attribute__((ext_vector_type(8)));

#define MDIM   12288
#define NCOL   8      // B*C_in valid columns
#define KTAPS  5
#define COUT   16
#define JC     64     // j-chunk (K dimension tile)
#define LDSS   68     // padded LDS row stride (floats) -> conflict-free strided reads
#define ROWS   32     // rows per block (2 waves x 16 rows)

// P0[m, i] = x[i, m]
__global__ __launch_bounds__(256) void sc_init_p0(const float* __restrict__ x,
                                                  float* __restrict__ P0) {
    int m = blockIdx.x * blockDim.x + threadIdx.x;
    if (m < MDIM) {
#pragma unroll
        for (int i = 0; i < NCOL; ++i)
            P0[m * NCOL + i] = x[i * MDIM + m];
    }
}

// Pout = L @ Pin   (L: [M,M] row-major, Pin/Pout: [M, 8])
// One wave computes a 16-row strip via V_WMMA_F32_16X16X4_F32 over the K dim.
__global__ __launch_bounds__(64) void sc_gemv_step(const float* __restrict__ Lm,
                                                   const float* __restrict__ Pin,
                                                   float* __restrict__ Pout) {
    __shared__ float tileA[ROWS * LDSS];

    const int t        = threadIdx.x;
    const int wave     = t >> 5;          // 0..1
    const int lane     = t & 31;
    const int halflane = lane & 15;
    const int koff     = (lane >> 4) * 2; // K offset per ISA A-layout: v0={K0|K2}, v1={K1|K3}
    const int row0     = blockIdx.x * ROWS;

    // cooperative tile-load mapping: 16 lanes x v4f cover one 64-float row
    const int ldRow = t >> 4;        // 0..3
    const int ldCol = (t & 15) * 4;  // 0..60

    v8f acc = {0.f, 0.f, 0.f, 0.f, 0.f, 0.f, 0.f, 0.f};

    for (int j0 = 0; j0 < MDIM; j0 += JC) {
        __syncthreads();
        // Stage L[row0:row0+32, j0:j0+64] into LDS. Non-temporal: L is
        // single-use this pass; keep L2 for the resident Pin vector block.
#pragma unroll
        for (int r = 0; r < ROWS; r += 4) {
            const v4f* src = reinterpret_cast<const v4f*>(
                &Lm[(size_t)(row0 + r + ldRow) * MDIM + (size_t)(j0 + ldCol)]);
            v4f v = __builtin_nontemporal_load(src);
            *reinterpret_cast<v4f*>(&tileA[(r + ldRow) * LDSS + ldCol]) = v;
        }
        __syncthreads();

        const int wrow = wave * 16;
#pragma unroll
        for (int jj = 0; jj < JC; jj += 4) {
            // A fragment (16x4 f32): lane holds A[halflane][koff], A[halflane][koff+1]
            v2f a;
            a.x = tileA[(wrow + halflane) * LDSS + jj + koff];
            a.y = tileA[(wrow + halflane) * LDSS + jj + koff + 1];

            // B fragment (4x16 f32): lane holds B[koff][n], B[koff+1][n], n = halflane.
            // Columns 8..15 are padding -> zero (cols are independent in D).
            float b0 = 0.f, b1 = 0.f;
            if (halflane < NCOL) {
                b0 = Pin[(size_t)(j0 + jj + koff)     * NCOL + halflane];
                b1 = Pin[(size_t)(j0 + jj + koff + 1) * NCOL + halflane];
            }
            v2f b;
            b.x = b0;
            b.y = b1;

            acc = __builtin_amdgcn_wmma_f32_16x16x4_f32(
                /*neg_a=*/false, a, /*neg_b=*/false, b,
                /*c_mod=*/(short)0, acc, /*reuse_a=*/false, /*reuse_b=*/false);
        }
    }

    // D layout: VGPR g -> rows {g | g+8} split across lane halves, N = halflane
    const int mrow = row0 + wave * 16 + ((lane >> 4) * 8);
    if (halflane < NCOL) {
#pragma unroll
        for (int g = 0; g < 8; ++g)
            Pout[(size_t)(mrow + g) * NCOL + halflane] = acc[g];
    }
}

// y[o, m] = bias[o] + sum_{i,k} theta[o,i,k] * P_k[m, i]
__global__ __launch_bounds__(256) void sc_combine(const float* __restrict__ P,
                                                  const float* __restrict__ theta,
                                                  const float* __restrict__ bias,
                                                  float* __restrict__ y) {
    int m = blockIdx.x * blockDim.x + threadIdx.x;
    if (m >= MDIM) return;

    float p[KTAPS * NCOL];
#pragma unroll
    for (int k = 0; k < KTAPS; ++k)
#pragma unroll
        for (int i = 0; i < NCOL; ++i)
            p[k * NCOL + i] = P[(size_t)k * MDIM * NCOL + (size_t)m * NCOL + i];

#pragma unroll
    for (int o = 0; o < COUT; ++o) {
        float acc = bias[o];
#pragma unroll
        for (int i = 0; i < NCOL; ++i)
#pragma unroll
            for (int k = 0; k < KTAPS; ++k)
                acc = fmaf(theta[(o * NCOL + i) * KTAPS + k], p[k * NCOL + i], acc);
        y[(size_t)o * MDIM + m] = acc;
    }
}

extern "C" void kernel_launch(void* const* d_in, const int* in_sizes, int n_in,
                              void* d_out, int out_size, void* d_ws, size_t ws_size,
                              hipStream_t stream) {
    const float* L     = (const float*)d_in[0];
    const float* x     = (const float*)d_in[1];
    const float* theta = (const float*)d_in[2];
    const float* bias  = (const float*)d_in[3];
    float* out = (float*)d_out;
    float* P   = (float*)d_ws;   // KTAPS * M * 8 floats (~2 MB)

    const size_t pstride = (size_t)MDIM * NCOL;

    sc_init_p0<<<(MDIM + 255) / 256, 256, 0, stream>>>(x, P);
    for (int k = 0; k < KTAPS - 1; ++k) {
        sc_gemv_step<<<MDIM / ROWS, 64, 0, stream>>>(L, P + k * pstride, P + (k + 1) * pstride);
    }
    sc_combine<<<(MDIM + 255) / 256, 256, 0, stream>>>(P, theta, bias, out);
}